// GCM_msf_41686952575686
// MI455X (gfx1250) — compile-verified
//
#include <hip/hip_runtime.h>
#include <hip/hip_bf16.h>
#include <math.h>

// ---------------------------------------------------------------------------
// MI455X (gfx1250, wave32) implementation of the multi-scale global-context
// module.  All GEMMs with M%16==0 (x->s conv, V projection, attention V*P,
// output-head convs) run on V_WMMA_F32_16X16X4_F32 (full fp32 precision).
// Each wave owns a 16x64 D tile (4 accumulators); B strides are template
// constants so the K-loop is pointer-increment + global_load clause + WMMA.
// Hot-path pointers are cast to address_space(1) so loads/stores lower to
// GLOBAL_* (LOADcnt/STOREcnt only) instead of FLAT_* (which also ties DScnt).
// ---------------------------------------------------------------------------

typedef float v2f __attribute__((ext_vector_type(2)));
typedef float v8f __attribute__((ext_vector_type(8)));

#define GAS __attribute__((address_space(1)))
typedef const GAS float* gcf_t;
typedef GAS float*       gf_t;

#define EPI_BIAS_RELU 0
#define EPI_GAMMA_RES 1
#define EPI_BIAS      2

// D[b][m][n] = epilogue( sum_k A[b][m][k] * B[b][k][n] )
//   A: row-major (row stride lda, batch stride a_bs)
//   B: element (k,n) at Bm + b*b_bs + k*BKS + n*BNS  (compile-time strides)
// One wave per (16 x 16*NT) tile of D; grid = (N/(16*NT), M/16, B).
template <int BKS, int BNS, int NT>
__global__ void wmma_gemm_kernel(
    const float* __restrict__ A, long a_bs, long lda,
    const float* __restrict__ Bm, long b_bs,
    const float* __restrict__ bias,
    const float* __restrict__ res, long res_bs,
    const float* __restrict__ gamma_ptr,
    float* __restrict__ D, long d_bs, long ldd,
    int K, int mode)
{
    const int mt = blockIdx.y;
    const int b  = blockIdx.z;
    const int lane = threadIdx.x & 31;
    const int hf   = lane >> 4;   // half-wave select (0: lanes 0-15, 1: 16-31)
    const int l    = lane & 15;

    // A 16x4 fragment: lane l holds row (mt*16+l); VGPR0/1 = K {0,1} for the
    // low half-wave, K {2,3} for the high half-wave.
    gcf_t Ap = (gcf_t)(A + b * a_bs + (long)(mt * 16 + l) * lda + hf * 2);

    // B 4x16 fragment mirrors C: VGPR0 = row K {0|2}, VGPR1 = row K {1|3};
    // lane l holds column (tile*16 + l).
    gcf_t Bp[NT];
    #pragma unroll
    for (int t = 0; t < NT; t++) {
        const int col = (blockIdx.x * NT + t) * 16 + l;
        Bp[t] = (gcf_t)(Bm + b * b_bs + (long)col * BNS + (long)(hf * 2) * BKS);
    }

    v8f acc[NT];
    #pragma unroll
    for (int t = 0; t < NT; t++) acc[t] = (v8f){0.f,0.f,0.f,0.f,0.f,0.f,0.f,0.f};

    const int steps = K >> 2;
    #pragma unroll 2
    for (int it = 0; it < steps; it++) {
        v2f av;
        av.x = Ap[0];
        av.y = Ap[1];
        v2f bv[NT];
        #pragma unroll
        for (int t = 0; t < NT; t++) {
            bv[t].x = Bp[t][0];
            bv[t].y = Bp[t][BKS];
        }
        #pragma unroll
        for (int t = 0; t < NT; t++) {
            acc[t] = __builtin_amdgcn_wmma_f32_16x16x4_f32(
                false, av, false, bv[t], (short)0, acc[t], false, false);
        }
        Ap += 4;
        #pragma unroll
        for (int t = 0; t < NT; t++) Bp[t] += 4 * BKS;
    }

    // C/D 16x16 layout: VGPR v -> row (v + 8*half), col = lane (mod 16).
    gf_t  Dp = (gf_t)(D + b * d_bs);
    if (mode == EPI_GAMMA_RES) {
        const float g = gamma_ptr[0];
        gcf_t Rp = (gcf_t)(res + b * res_bs);
        #pragma unroll
        for (int t = 0; t < NT; t++) {
            const int n = (blockIdx.x * NT + t) * 16 + l;
            #pragma unroll
            for (int vi = 0; vi < 8; vi++) {
                const int m = mt * 16 + vi + 8 * hf;
                Dp[(long)m * ldd + n] = g * acc[t][vi] + Rp[(long)m * ldd + n];
            }
        }
    } else {
        #pragma unroll
        for (int t = 0; t < NT; t++) {
            const int n = (blockIdx.x * NT + t) * 16 + l;
            #pragma unroll
            for (int vi = 0; vi < 8; vi++) {
                const int m = mt * 16 + vi + 8 * hf;
                float val = acc[t][vi] + (bias ? bias[m] : 0.f);
                if (mode == EPI_BIAS_RELU) val = fmaxf(val, 0.f);
                Dp[(long)m * ldd + n] = val;
            }
        }
    }
}

// Adaptive average pool: 35 regions (1x1 + 3x3 + 5x5, overlapping bins).
// p layout: (B, 512, 35).  grid = (35, B), block = 256.
__global__ void pool_kernel(const float* __restrict__ x, float* __restrict__ p,
                            int H, int W)
{
    const int r = blockIdx.x, b = blockIdx.y;
    int ps, ri, rj;
    if (r == 0)      { ps = 1; ri = 0;            rj = 0;            }
    else if (r < 10) { ps = 3; ri = (r - 1) / 3;  rj = (r - 1) % 3;  }
    else             { ps = 5; ri = (r - 10) / 5; rj = (r - 10) % 5; }
    const int hs = ri * H / ps, he = ((ri + 1) * H + ps - 1) / ps;
    const int ws0 = rj * W / ps, we = ((rj + 1) * W + ps - 1) / ps;
    const float inv = 1.f / (float)((he - hs) * (we - ws0));
    for (int c = threadIdx.x; c < 512; c += blockDim.x) {
        gcf_t xp = (gcf_t)(x + ((long)b * 512 + c) * H * W);
        float sum = 0.f;
        for (int y = hs; y < he; y++)
            for (int xx = ws0; xx < we; xx++)
                sum += xp[y * W + xx];
        p[((long)b * 512 + c) * 35 + r] = sum * inv;
    }
}

// Small 1x1 conv (VALU): out[b][co][n] = act(sum_ci w[co][ci]*in(ci,n) + bias)
// in element (ci,n) at  in + b*in_bs + ci*in_cs + n   (n-stride always 1).
__global__ void conv1x1_kernel(const float* __restrict__ in,
                               const float* __restrict__ w,
                               const float* __restrict__ bias,
                               float* __restrict__ out,
                               int Cin, int Cout, int N,
                               long in_cs, long in_bs, long out_bs, int do_relu)
{
    const long idx = (long)blockIdx.x * blockDim.x + threadIdx.x;
    if (idx >= (long)Cout * N) return;
    const int b = blockIdx.y;
    const int co = (int)(idx / N);
    const int n  = (int)(idx % N);
    gcf_t ip = (gcf_t)(in + b * in_bs + n);
    gcf_t wp = (gcf_t)(w + (long)co * Cin);
    float acc = bias ? bias[co] : 0.f;
    for (int ci = 0; ci < Cin; ci++) acc += wp[ci] * ip[(long)ci * in_cs];
    if (do_relu) acc = fmaxf(acc, 0.f);
    out[b * out_bs + (long)co * N + n] = acc;
}

// Row softmax of Q^T K (K-dim = 8).  One block per attention row m.
// Row (2304 floats) staged in LDS; P[b][m][n] = softmax_n(sum_c q[c,m]k[c,n]).
__global__ void attn_softmax_kernel(const float* __restrict__ q,
                                    const float* __restrict__ k,
                                    float* __restrict__ P, int N)
{
    const int m = blockIdx.x, b = blockIdx.y;
    __shared__ float row[2304];
    __shared__ float red[256];
    const int tid = threadIdx.x;
    gcf_t qb = (gcf_t)(q + (long)b * 8 * N);
    gcf_t kb = (gcf_t)(k + (long)b * 8 * N);
    float qm[8];
    #pragma unroll
    for (int c = 0; c < 8; c++) qm[c] = qb[c * N + m];

    float lmax = -3.4e38f;
    for (int n = tid; n < N; n += 256) {
        float s = 0.f;
        #pragma unroll
        for (int c = 0; c < 8; c++) s += qm[c] * kb[c * N + n];
        row[n] = s;
        lmax = fmaxf(lmax, s);
    }
    red[tid] = lmax; __syncthreads();
    for (int o = 128; o > 0; o >>= 1) {
        if (tid < o) red[tid] = fmaxf(red[tid], red[tid + o]);
        __syncthreads();
    }
    const float gmax = red[0]; __syncthreads();

    float lsum = 0.f;
    for (int n = tid; n < N; n += 256) {
        const float e = expf(row[n] - gmax);
        row[n] = e; lsum += e;
    }
    red[tid] = lsum; __syncthreads();
    for (int o = 128; o > 0; o >>= 1) {
        if (tid < o) red[tid] += red[tid + o];
        __syncthreads();
    }
    const float inv = 1.f / red[0];
    gf_t Pr = (gf_t)(P + ((long)b * N + m) * N);
    for (int n = tid; n < N; n += 256) Pr[n] = row[n] * inv;
}

// Generic bilinear resize, NCHW.  grid.x covers C*OH*OW, grid.y = B.
__global__ void bilinear_kernel(const float* __restrict__ in,
                                float* __restrict__ out,
                                int C, int IH, int IW, int OH, int OW,
                                int align, long in_bs, long out_bs)
{
    const long idx = (long)blockIdx.x * blockDim.x + threadIdx.x;
    const long total = (long)C * OH * OW;
    if (idx >= total) return;
    const int b  = blockIdx.y;
    const int ox = (int)(idx % OW);
    const int oy = (int)((idx / OW) % OH);
    const int c  = (int)(idx / ((long)OW * OH));
    float fy, fx;
    if (align) {
        fy = (OH > 1) ? oy * (float)(IH - 1) / (float)(OH - 1) : 0.f;
        fx = (OW > 1) ? ox * (float)(IW - 1) / (float)(OW - 1) : 0.f;
    } else {
        fy = fminf(fmaxf((oy + 0.5f) * (float)IH / (float)OH - 0.5f, 0.f), (float)(IH - 1));
        fx = fminf(fmaxf((ox + 0.5f) * (float)IW / (float)OW - 0.5f, 0.f), (float)(IW - 1));
    }
    const int y0 = (int)floorf(fy); const int y1 = min(y0 + 1, IH - 1);
    const int x0 = (int)floorf(fx); const int x1 = min(x0 + 1, IW - 1);
    const float wy = fy - (float)y0, wx = fx - (float)x0;
    gcf_t ip = (gcf_t)(in + (long)b * in_bs + (long)c * IH * IW);
    const float a  = ip[y0 * IW + x0], bb = ip[y0 * IW + x1];
    const float cc = ip[y1 * IW + x0], dd = ip[y1 * IW + x1];
    const float r0 = a * (1.f - wx) + bb * wx;
    const float r1 = cc * (1.f - wx) + dd * wx;
    gf_t op = (gf_t)(out + (long)b * out_bs);
    op[((long)c * OH + oy) * OW + ox] = r0 * (1.f - wy) + r1 * wy;
}

// Global average pool of the four (B,64,N) fields.  grid = (64, B, 4).
__global__ void gap_kernel(const float* __restrict__ s0, const float* __restrict__ s1,
                           const float* __restrict__ s2, const float* __restrict__ s3,
                           float* __restrict__ gapo, int N)
{
    const int c = blockIdx.x, b = blockIdx.y, si = blockIdx.z;
    const float* sp = (si == 0) ? s0 : (si == 1) ? s1 : (si == 2) ? s2 : s3;
    gcf_t p = (gcf_t)(sp + ((long)b * 64 + c) * N);
    __shared__ float red[256];
    const int tid = threadIdx.x;
    float ls = 0.f;
    for (int n = tid; n < N; n += 256) ls += p[n];
    red[tid] = ls; __syncthreads();
    for (int o = 128; o > 0; o >>= 1) {
        if (tid < o) red[tid] += red[tid + o];
        __syncthreads();
    }
    if (tid == 0) gapo[((long)si * gridDim.y + b) * 64 + c] = red[0] / (float)N;
}

// SE matvecs + sigmoid + softmax over the 4 branch weights.  grid=(B), blk=64.
__global__ void se_kernel(const float* __restrict__ gapb,
                          const float* __restrict__ w1, const float* __restrict__ w2,
                          const float* __restrict__ w3, const float* __restrict__ w4,
                          float* __restrict__ wgt, int B)
{
    const int b = blockIdx.x, c = threadIdx.x;
    const float* g0 = gapb + ((long)0 * B + b) * 64;  // gap(gc ps=1)
    const float* g1 = gapb + ((long)1 * B + b) * 64;  // gap(gc ps=3)
    const float* g2 = gapb + ((long)2 * B + b) * 64;  // gap(gc ps=5)
    const float* g3 = gapb + ((long)3 * B + b) * 64;  // gap(sa)
    float y0 = 0.f, y1 = 0.f, y2 = 0.f, y3 = 0.f;
    for (int i = 0; i < 64; i++) {
        y0 += w1[c * 64 + i] * g3[i];   // w_se1 @ gap(sa)
        y1 += w2[c * 64 + i] * g2[i];   // w_se2 @ gap(gcs[2])
        y2 += w3[c * 64 + i] * g1[i];   // w_se3 @ gap(gcs[1])
        y3 += w4[c * 64 + i] * g0[i];   // w_se4 @ gap(gcs[0])
    }
    const float s0 = 1.f / (1.f + expf(-y0));
    const float s1 = 1.f / (1.f + expf(-y1));
    const float s2 = 1.f / (1.f + expf(-y2));
    const float s3 = 1.f / (1.f + expf(-y3));
    const float mx = fmaxf(fmaxf(s0, s1), fmaxf(s2, s3));
    const float e0 = expf(s0 - mx), e1 = expf(s1 - mx);
    const float e2 = expf(s2 - mx), e3 = expf(s3 - mx);
    const float inv = 1.f / (e0 + e1 + e2 + e3);
    float* wp = wgt + ((long)b * 64 + c) * 4;
    wp[0] = e0 * inv; wp[1] = e1 * inv; wp[2] = e2 * inv; wp[3] = e3 * inv;
}

// fused[b][cc][n] = wgt[b][cc&63][cc>>6] * branch(cc>>6)[b][cc&63][n]
__global__ void fuse_kernel(const float* __restrict__ up0, const float* __restrict__ up1,
                            const float* __restrict__ up2, const float* __restrict__ sa,
                            const float* __restrict__ wgt, float* __restrict__ fused,
                            int N)
{
    const long idx = (long)blockIdx.x * blockDim.x + threadIdx.x;
    if (idx >= (long)256 * N) return;
    const int b  = blockIdx.y;
    const int n  = (int)(idx % N);
    const int cc = (int)(idx / N);
    const int blk = cc >> 6, c = cc & 63;
    const float* sp = (blk == 0) ? up0 : (blk == 1) ? up1 : (blk == 2) ? up2 : sa;
    const float w = wgt[((long)b * 64 + c) * 4 + blk];
    gcf_t ip = (gcf_t)(sp + ((long)b * 64 + c) * N);
    gf_t  op = (gf_t)(fused + ((long)b * 256 + cc) * N);
    op[n] = w * ip[n];
}

extern "C" void kernel_launch(void* const* d_in, const int* in_sizes, int n_in,
                              void* d_out, int out_size, void* d_ws, size_t ws_size,
                              hipStream_t stream)
{
    (void)in_sizes; (void)n_in; (void)out_size; (void)ws_size;
    const int B = 2, Cin = 512, C = 64, H = 48, W = 48, N = H * W; // N = 2304
    const int NT = 4;                       // N tiles per wave in WMMA GEMMs

    const float* x      = (const float*)d_in[0];
    const float* w_gc0  = (const float*)d_in[1];  const float* b_gc0 = (const float*)d_in[2];
    const float* w_gc1  = (const float*)d_in[3];  const float* b_gc1 = (const float*)d_in[4];
    const float* w_gc2  = (const float*)d_in[5];  const float* b_gc2 = (const float*)d_in[6];
    const float* w_gc3  = (const float*)d_in[7];  const float* b_gc3 = (const float*)d_in[8];
    const float* w_q    = (const float*)d_in[9];  const float* b_q   = (const float*)d_in[10];
    const float* w_k    = (const float*)d_in[11]; const float* b_k   = (const float*)d_in[12];
    const float* w_v    = (const float*)d_in[13]; const float* b_v   = (const float*)d_in[14];
    const float* gma    = (const float*)d_in[15];
    const float* w_se1  = (const float*)d_in[16]; const float* w_se2 = (const float*)d_in[17];
    const float* w_se3  = (const float*)d_in[18]; const float* w_se4 = (const float*)d_in[19];
    const float* w_out0 = (const float*)d_in[20]; const float* b_out0 = (const float*)d_in[21];
    const float* w_out1 = (const float*)d_in[22]; const float* b_out1 = (const float*)d_in[23];
    const float* w_out2 = (const float*)d_in[24]; const float* b_out2 = (const float*)d_in[25];
    const float* w_out3 = (const float*)d_in[26]; const float* b_out3 = (const float*)d_in[27];

    // ---- workspace partition (floats) -------------------------------------
    float* ws = (float*)d_ws;
    size_t off = 0;
    auto take = [&](size_t nf) { float* r = ws + off; off += nf; return r; };
    float* p    = take((size_t)B * 512 * 35);    // pooled (B,512,35)
    float* gcs1 = take((size_t)B * C * 1);       // gc conv results (small)
    float* gcs3 = take((size_t)B * C * 9);
    float* gcs5 = take((size_t)B * C * 25);
    float* s    = take((size_t)B * C * N);       // relu(conv(x, w_gc3))
    float* q    = take((size_t)B * 8 * N);
    float* kk   = take((size_t)B * 8 * N);
    float* v    = take((size_t)B * C * N);
    float* P    = take((size_t)B * N * N);       // 42.5 MB, lives in L2
    float* sa   = take((size_t)B * C * N);
    float* up0  = take((size_t)B * C * N);
    float* up1  = take((size_t)B * C * N);
    float* up2  = take((size_t)B * C * N);
    float* gapb = take((size_t)4 * B * C);
    float* wgt  = take((size_t)B * C * 4);
    float* fused= take((size_t)B * 256 * N);
    float* y0   = take((size_t)B * 256 * N);
    float* y1   = take((size_t)B * 128 * N);
    float* y2   = take((size_t)B * 64 * N);
    float* y3   = take((size_t)B * 64 * N);

    const unsigned gx = (unsigned)(N / (16 * NT));   // 36 N-tiles per row

    // ---- 1) adaptive pools ------------------------------------------------
    pool_kernel<<<dim3(35, B), 256, 0, stream>>>(x, p, H, W);

    // ---- 2) gc branch convs (tiny, VALU) ----------------------------------
    conv1x1_kernel<<<dim3(1, B), 256, 0, stream>>>(p + 0,  w_gc0, b_gc0, gcs1,
        Cin, C, 1,  35L, (long)512 * 35, (long)C * 1, 1);
    conv1x1_kernel<<<dim3((C * 9 + 255) / 256, B), 256, 0, stream>>>(p + 1,  w_gc1, b_gc1, gcs3,
        Cin, C, 9,  35L, (long)512 * 35, (long)C * 9, 1);
    conv1x1_kernel<<<dim3((C * 25 + 255) / 256, B), 256, 0, stream>>>(p + 10, w_gc2, b_gc2, gcs5,
        Cin, C, 25, 35L, (long)512 * 35, (long)C * 25, 1);

    // ---- 3) s = relu(W_gc3 x + b)  [WMMA f32, M=64 K=512 N=2304] ----------
    wmma_gemm_kernel<2304, 1, 4><<<dim3(gx, C / 16, B), 32, 0, stream>>>(
        w_gc3, 0L, (long)Cin,
        x, (long)Cin * N,
        b_gc3, nullptr, 0L, gma,
        s, (long)C * N, (long)N, Cin, EPI_BIAS_RELU);

    // ---- 4) q, k projections (M=8: VALU); v (M=64): WMMA ------------------
    conv1x1_kernel<<<dim3((8 * N + 255) / 256, B), 256, 0, stream>>>(s, w_q, b_q, q,
        C, 8, N, (long)N, (long)C * N, (long)8 * N, 0);
    conv1x1_kernel<<<dim3((8 * N + 255) / 256, B), 256, 0, stream>>>(s, w_k, b_k, kk,
        C, 8, N, (long)N, (long)C * N, (long)8 * N, 0);
    wmma_gemm_kernel<2304, 1, 4><<<dim3(gx, C / 16, B), 32, 0, stream>>>(
        w_v, 0L, (long)C,
        s, (long)C * N,
        b_v, nullptr, 0L, gma,
        v, (long)C * N, (long)N, C, EPI_BIAS);

    // ---- 5) attention scores + row softmax --------------------------------
    attn_softmax_kernel<<<dim3(N, B), 256, 0, stream>>>(q, kk, P, N);

    // ---- 6) sa = gamma * (V P^T) + s  [WMMA f32, M=64 K=2304 N=2304] ------
    // B(k,n) = P[b][n][k]  ->  k-stride 1 (contiguous pair load), n-stride N.
    wmma_gemm_kernel<1, 2304, 4><<<dim3(gx, C / 16, B), 32, 0, stream>>>(
        v, (long)C * N, (long)N,
        P, (long)N * N,
        nullptr, s, (long)C * N, gma,
        sa, (long)C * N, (long)N, N, EPI_GAMMA_RES);

    // ---- 7) upsample gc branches to 48x48 (align_corners=True) ------------
    const unsigned gUp = (unsigned)((C * N + 255) / 256);
    bilinear_kernel<<<dim3(gUp, B), 256, 0, stream>>>(gcs1, up0, C, 1, 1, H, W, 1,
        (long)C * 1,  (long)C * N);
    bilinear_kernel<<<dim3(gUp, B), 256, 0, stream>>>(gcs3, up1, C, 3, 3, H, W, 1,
        (long)C * 9,  (long)C * N);
    bilinear_kernel<<<dim3(gUp, B), 256, 0, stream>>>(gcs5, up2, C, 5, 5, H, W, 1,
        (long)C * 25, (long)C * N);

    // ---- 8) GAP + SE weights + fuse ---------------------------------------
    gap_kernel<<<dim3(C, B, 4), 256, 0, stream>>>(up0, up1, up2, sa, gapb, N);
    se_kernel<<<dim3(B), 64, 0, stream>>>(gapb, w_se1, w_se2, w_se3, w_se4, wgt, B);
    fuse_kernel<<<dim3((unsigned)((256 * N + 255) / 256), B), 256, 0, stream>>>(
        up0, up1, up2, sa, wgt, fused, N);

    // ---- 9) output-head convs [WMMA f32, K=256] ---------------------------
    wmma_gemm_kernel<2304, 1, 4><<<dim3(gx, 256 / 16, B), 32, 0, stream>>>(
        w_out0, 0L, 256L, fused, (long)256 * N,
        b_out0, nullptr, 0L, gma, y0, (long)256 * N, (long)N, 256, EPI_BIAS_RELU);
    wmma_gemm_kernel<2304, 1, 4><<<dim3(gx, 128 / 16, B), 32, 0, stream>>>(
        w_out1, 0L, 256L, fused, (long)256 * N,
        b_out1, nullptr, 0L, gma, y1, (long)128 * N, (long)N, 256, EPI_BIAS_RELU);
    wmma_gemm_kernel<2304, 1, 4><<<dim3(gx, 64 / 16, B), 32, 0, stream>>>(
        w_out2, 0L, 256L, fused, (long)256 * N,
        b_out2, nullptr, 0L, gma, y2, (long)64 * N, (long)N, 256, EPI_BIAS_RELU);
    wmma_gemm_kernel<2304, 1, 4><<<dim3(gx, 64 / 16, B), 32, 0, stream>>>(
        w_out3, 0L, 256L, fused, (long)256 * N,
        b_out3, nullptr, 0L, gma, y3, (long)64 * N, (long)N, 256, EPI_BIAS_RELU);

    // ---- 10) final bilinear upsamples into d_out (align_corners=False) ----
    float* out = (float*)d_out;
    const size_t o0 = 0;
    const size_t o1 = o0 + (size_t)B * 256 * 96 * 96;
    const size_t o2 = o1 + (size_t)B * 128 * 192 * 192;
    const size_t o3 = o2 + (size_t)B * 64 * 384 * 384;
    bilinear_kernel<<<dim3((unsigned)(((size_t)256 * 96 * 96 + 255) / 256), B), 256, 0, stream>>>(
        y0, out + o0, 256, H, W, 96, 96, 0, (long)256 * N, (long)256 * 96 * 96);
    bilinear_kernel<<<dim3((unsigned)(((size_t)128 * 192 * 192 + 255) / 256), B), 256, 0, stream>>>(
        y1, out + o1, 128, H, W, 192, 192, 0, (long)128 * N, (long)128 * 192 * 192);
    bilinear_kernel<<<dim3((unsigned)(((size_t)64 * 384 * 384 + 255) / 256), B), 256, 0, stream>>>(
        y2, out + o2, 64, H, W, 384, 384, 0, (long)64 * N, (long)64 * 384 * 384);
    bilinear_kernel<<<dim3((unsigned)(((size_t)64 * 768 * 768 + 255) / 256), B), 256, 0, stream>>>(
        y3, out + o3, 64, H, W, 768, 768, 0, (long)64 * N, (long)64 * 768 * 768);
}